// Simple_Model_63582695850427
// MI455X (gfx1250) — compile-verified
//
#include <hip/hip_runtime.h>
#include <hip/hip_bf16.h>
#include <math.h>

// Sizes from the reference
#define BB    8192
#define NN    16
#define NIN   512
#define NH    256
#define BPB   8          // batches per block (register-level W_gcn reuse factor)
#define KC    128        // K-chunk staged in LDS

typedef float v2f __attribute__((ext_vector_type(2)));
typedef float v8f __attribute__((ext_vector_type(8)));

// D(16x16,f32) = A(16x4,f32) x B(4x16,f32) + C   -- full fp32 WMMA
__device__ __forceinline__ v8f wmma4(v2f a, v2f b, v8f c) {
    return __builtin_amdgcn_wmma_f32_16x16x4_f32(
        /*neg_a=*/false, a, /*neg_b=*/false, b,
        /*c_mod=*/(short)0, c, /*reuse_a=*/false, /*reuse_b=*/false);
}

// ---------------------------------------------------------------------------
// Kernel 1: 8 batches per block. For each batch:
//   seq_fts = seq1_b @ W_gcn ; h1 = adj_b @ seq_fts + b ; PReLU ;
//   c = mean(h1[0:15,:]) ; hmv = h1[15,:]
// 512 threads = 16 waves; wave w owns N-cols [16w,16w+16) for ALL 8 batches,
// so each W_gcn B-fragment (2 VGPRs) feeds 8 WMMAs (8x L2 traffic reduction).
// Dynamic LDS buffer (128 KB) reused: A-chunks -> seq_fts -> h1.
// ---------------------------------------------------------------------------
__global__ __launch_bounds__(512)
void gcn_kernel(const float* __restrict__ seq1, const float* __restrict__ adj,
                const float* __restrict__ Wg,   const float* __restrict__ bg,
                const float* __restrict__ prelu_a,
                float* __restrict__ c_out, float* __restrict__ hmv_out)
{
    extern __shared__ float SH[];      // 32768 floats = 128 KB

    const int b8   = blockIdx.x * BPB;
    const int t    = threadIdx.x;      // 0..511
    const int wave = t >> 5;           // 0..15
    const int lane = t & 31;
    const int l15  = lane & 15;        // A: row M ; B: col N ; C/D: col N
    const int kh   = (lane >> 4) * 2;  // K sub-offset per half-wave (0 or 2)
    const int row0 = (lane >> 4) * 8;  // C/D: M base per half-wave
    const int nb   = wave * 16;        // N-tile base

    // ---- GEMM1: acc[m] = seq1[b8+m](16x512) @ W_gcn[:, nb:nb+16] ----
    v8f acc[BPB];
    #pragma unroll
    for (int m = 0; m < BPB; ++m) acc[m] = (v8f){0.f,0.f,0.f,0.f,0.f,0.f,0.f,0.f};

    for (int kc = 0; kc < NIN; kc += KC) {
        // Stage A-chunk: SH[m][row][kk] for kk in [kc,kc+KC), 16384 floats
        #pragma unroll
        for (int i = 0; i < (BPB * NN * KC) / 512; ++i) {
            const int id  = t + i * 512;
            const int m   = id / (NN * KC);
            const int rem = id % (NN * KC);
            const int row = rem / KC;
            const int kk  = rem % KC;
            SH[id] = seq1[(size_t)(b8 + m) * (NN * NIN) + row * NIN + kc + kk];
        }
        __syncthreads();

        for (int k = 0; k < KC; k += 4) {
            v2f w;   // one B-fragment, reused by 8 WMMAs
            w.x = Wg[(size_t)(kc + k + kh)     * NH + nb + l15];
            w.y = Wg[(size_t)(kc + k + kh + 1) * NH + nb + l15];
            #pragma unroll
            for (int m = 0; m < BPB; ++m) {
                v2f a;
                const int base = m * (NN * KC) + l15 * KC + k + kh;
                a.x = SH[base];
                a.y = SH[base + 1];
                acc[m] = wmma4(a, w, acc[m]);
            }
        }
        __syncthreads();   // all waves done reading this A-chunk
    }

    // ---- Stage seq_fts for all 8 batches: SH[m][16][256] (128 KB) ----
    #pragma unroll
    for (int m = 0; m < BPB; ++m)
        #pragma unroll
        for (int v = 0; v < 8; ++v)
            SH[m * (NN * NH) + (row0 + v) * NH + nb + l15] = acc[m][v];
    __syncthreads();

    // ---- GEMM2: h[m] = adj[b8+m](16x16) @ seq_fts[m][:, nb:nb+16] ----
    v8f h[BPB];
    #pragma unroll
    for (int m = 0; m < BPB; ++m) h[m] = (v8f){0.f,0.f,0.f,0.f,0.f,0.f,0.f,0.f};

    #pragma unroll
    for (int kk = 0; kk < NN; kk += 4) {
        #pragma unroll
        for (int m = 0; m < BPB; ++m) {
            v2f a, s;
            const float* ab = adj + (size_t)(b8 + m) * (NN * NN);
            a.x = ab[l15 * NN + kk + kh];
            a.y = ab[l15 * NN + kk + kh + 1];
            s.x = SH[m * (NN * NH) + (kk + kh)     * NH + nb + l15];
            s.y = SH[m * (NN * NH) + (kk + kh + 1) * NH + nb + l15];
            h[m] = wmma4(a, s, h[m]);
        }
    }
    __syncthreads();   // all waves done reading seq_fts before H1 overwrite

    // ---- Fused bias + PReLU, stage h1 into the same buffer ----
    const float bias  = bg[nb + l15];
    const float alpha = prelu_a[0];
    #pragma unroll
    for (int m = 0; m < BPB; ++m)
        #pragma unroll
        for (int v = 0; v < 8; ++v) {
            float x = h[m][v] + bias;
            x = (x >= 0.f) ? x : alpha * x;
            SH[m * (NN * NH) + (row0 + v) * NH + nb + l15] = x;
        }
    __syncthreads();

    // ---- c = mean rows 0..14, hmv = row 15 : 2048 columns, 4 per thread ----
    #pragma unroll
    for (int j = 0; j < (BPB * NH) / 512; ++j) {
        const int id  = t + j * 512;
        const int m   = id >> 8;          // batch within block
        const int col = id & (NH - 1);
        float s = 0.f;
        #pragma unroll
        for (int r = 0; r < NN - 1; ++r) s += SH[m * (NN * NH) + r * NH + col];
        c_out  [(size_t)(b8 + m) * NH + col] = s * (1.0f / (float)(NN - 1));
        hmv_out[(size_t)(b8 + m) * NH + col] = SH[m * (NN * NH) + (NN - 1) * NH + col];
    }
}

// ---------------------------------------------------------------------------
// Kernel 2: P = C(8192x256) @ W_bil^T(256x256) -> P[b][i] = sum_j c[b][j]W[i][j]
// 512 blocks x 512 threads; block = 16-row M-tile, wave w = N-tile w
// ---------------------------------------------------------------------------
__global__ __launch_bounds__(512)
void bilin_gemm(const float* __restrict__ C, const float* __restrict__ Wb,
                float* __restrict__ P)
{
    __shared__ float Cs[16 * NH];    // 16 KB
    const int mb   = blockIdx.x * 16;
    const int t    = threadIdx.x;
    const int wave = t >> 5;
    const int lane = t & 31;
    const int l15  = lane & 15;
    const int kh   = (lane >> 4) * 2;
    const int row0 = (lane >> 4) * 8;
    const int nb   = wave * 16;

    #pragma unroll
    for (int i = 0; i < 8; ++i) Cs[i * 512 + t] = C[(size_t)mb * NH + i * 512 + t];
    __syncthreads();

    v8f acc = {0.f,0.f,0.f,0.f,0.f,0.f,0.f,0.f};
    #pragma unroll 4
    for (int k = 0; k < NH; k += 4) {
        v2f a, w;
        a.x = Cs[l15 * NH + k + kh];
        a.y = Cs[l15 * NH + k + kh + 1];
        // B[K=j][N=i] = W_bil[i][j]  (row-major W_bil, hot in L2)
        w.x = Wb[(size_t)(nb + l15) * NH + k + kh];
        w.y = Wb[(size_t)(nb + l15) * NH + k + kh + 1];
        acc = wmma4(a, w, acc);
    }
    #pragma unroll
    for (int v = 0; v < 8; ++v)
        P[(size_t)(mb + row0 + v) * NH + nb + l15] = acc[v];
}

// ---------------------------------------------------------------------------
// Kernel 3: one wave32 per batch: pos/neg bilinear scores + hinge loss
// out[0:B) = pos logits, out[B:2B) = neg logits, out[2B:3B) = loss2
// ---------------------------------------------------------------------------
__global__ __launch_bounds__(256)
void score_kernel(const float* __restrict__ c, const float* __restrict__ hmv,
                  const float* __restrict__ P, const float* __restrict__ b_bil,
                  float* __restrict__ out)
{
    const int b    = (blockIdx.x * blockDim.x + threadIdx.x) >> 5;
    const int lane = threadIdx.x & 31;
    if (b >= BB) return;
    const int idx = (b == 0) ? (BB - 2) : (b - 1);   // c_mi[b] = c[idx]

    const float* hb = hmv + (size_t)b   * NH;
    const float* cb = c   + (size_t)b   * NH;
    const float* cn = c   + (size_t)idx * NH;
    const float* pb = P   + (size_t)b   * NH;
    const float* pn = P   + (size_t)idx * NH;

    float sp = 0.f, sn = 0.f, dp2 = 0.f, dn2 = 0.f;
    #pragma unroll
    for (int i = lane; i < NH; i += 32) {
        const float h = hb[i];
        sp += h * pb[i];
        sn += h * pn[i];
        const float dp = h - cb[i] + 1e-6f;
        const float dn = h - cn[i] + 1e-6f;
        dp2 += dp * dp;
        dn2 += dn * dn;
    }
    #pragma unroll
    for (int off = 16; off >= 1; off >>= 1) {
        sp  += __shfl_down(sp,  off, 32);
        sn  += __shfl_down(sn,  off, 32);
        dp2 += __shfl_down(dp2, off, 32);
        dn2 += __shfl_down(dn2, off, 32);
    }
    if (lane == 0) {
        const float bb = b_bil[0];
        out[b]          = sp + bb;
        out[BB + b]     = sn + bb;
        const float l   = sqrtf(dp2) - sqrtf(dn2) + 0.5f;
        out[2 * BB + b] = fmaxf(l, 0.f);
    }
}

extern "C" void kernel_launch(void* const* d_in, const int* in_sizes, int n_in,
                              void* d_out, int out_size, void* d_ws, size_t ws_size,
                              hipStream_t stream)
{
    (void)in_sizes; (void)n_in; (void)out_size; (void)ws_size;

    const float* seq1    = (const float*)d_in[0];  // (B,16,512)
    const float* adj     = (const float*)d_in[1];  // (B,16,16)
    const float* W_gcn   = (const float*)d_in[2];  // (512,256)
    const float* b_gcn   = (const float*)d_in[3];  // (256,)
    const float* prelu_a = (const float*)d_in[4];  // scalar
    const float* W_bil   = (const float*)d_in[5];  // (256,256)
    const float* b_bil   = (const float*)d_in[6];  // scalar
    float* out = (float*)d_out;                    // 3*B floats

    // Workspace layout (needs 24 MB): c | hmv | P, each B x 256 fp32
    float* c_ws   = (float*)d_ws;
    float* hmv_ws = c_ws   + (size_t)BB * NH;
    float* P_ws   = hmv_ws + (size_t)BB * NH;

    const size_t lds_bytes = (size_t)BPB * NN * NH * sizeof(float);  // 128 KB

    gcn_kernel<<<BB / BPB, 512, lds_bytes, stream>>>(seq1, adj, W_gcn, b_gcn,
                                                     prelu_a, c_ws, hmv_ws);
    bilin_gemm<<<BB / 16, 512, 0, stream>>>(c_ws, W_bil, P_ws);
    score_kernel<<<(BB * 32) / 256, 256, 0, stream>>>(c_ws, hmv_ws, P_ws, b_bil, out);
}